// PatchLevelModeling_48868137894388
// MI455X (gfx1250) — compile-verified
//
#include <hip/hip_runtime.h>
#include <cmath>

typedef __attribute__((ext_vector_type(2))) float v2f;
typedef __attribute__((ext_vector_type(8))) float v8f;

#define B_    2
#define C_    96
#define ZD    64
#define HD    64
#define WD    64
#define L_    4096      // 16*16*16 tokens
#define DI    192       // d_inner
#define DS    8         // d_state
#define DTR   6         // dt_rank
#define NTOK  (B_*L_)   // 8192
#define CHUNK 64
#define NCH   (L_/CHUNK) // 64 chunks

// ---------------------------------------------------------------- pool 4x4x4
__global__ void k_pool(const float* __restrict__ x, float* __restrict__ xn) {
    int tid = blockIdx.x * blockDim.x + threadIdx.x;   // (b, c, l) with l fast
    if (tid >= B_ * C_ * L_) return;
    int l = tid % L_;
    int c = (tid / L_) % C_;
    int b = tid / (L_ * C_);
    int nw = l & 15, nh = (l >> 4) & 15, nz = l >> 8;
    const float* xp = x + ((((size_t)(b * C_ + c) * ZD + nz * 4) * HD + nh * 4) * WD + nw * 4);
    float s = 0.f;
    #pragma unroll
    for (int dz = 0; dz < 4; ++dz)
        #pragma unroll
        for (int dh = 0; dh < 4; ++dh) {
            const float* r = xp + (size_t)dz * HD * WD + dh * WD;
            s += r[0] + r[1] + r[2] + r[3];
        }
    xn[(size_t)(b * L_ + l) * C_ + c] = s * (1.f / 64.f);
}

// ------------------------------------------------------- layernorm (in place)
__global__ void k_ln(float* __restrict__ xn, const float* __restrict__ lnw,
                     const float* __restrict__ lnb) {
    int m = blockIdx.x * blockDim.x + threadIdx.x;
    if (m >= NTOK) return;
    float* row = xn + (size_t)m * C_;
    float s = 0.f, s2 = 0.f;
    for (int c = 0; c < C_; ++c) { float v = row[c]; s += v; s2 += v * v; }
    float mu  = s * (1.f / C_);
    float var = s2 * (1.f / C_) - mu * mu;
    float inv = rsqrtf(var + 1e-5f);
    for (int c = 0; c < C_; ++c)
        row[c] = (row[c] - mu) * inv * lnw[c] + lnb[c];
}

// ------------------------------------------- fp32 WMMA 16x16 tile (K step 4)
// Out[m,n] = sum_k A[m,k] * W[n,k]   (A row-major [M,lda], W row-major [N,ldw])
__device__ __forceinline__ void wmma_tile_f32(
        const float* __restrict__ A, int lda,
        const float* __restrict__ W, int ldw,
        float* __restrict__ Out, int ldo,
        int m0, int n0, int K, int lane) {
    int row = lane & 15;           // M index for A-frag / N index for B-frag
    int kh  = (lane >> 4) << 1;    // 0 or 2
    v8f acc = {0.f, 0.f, 0.f, 0.f, 0.f, 0.f, 0.f, 0.f};
    const float* ap = A + (size_t)(m0 + row) * lda + kh;
    const float* bp = W + (size_t)(n0 + row) * ldw + kh;
    for (int k0 = 0; k0 < K; k0 += 4) {
        v2f a, b;
        a.x = ap[k0]; a.y = ap[k0 + 1];
        b.x = bp[k0]; b.y = bp[k0 + 1];
        acc = __builtin_amdgcn_wmma_f32_16x16x4_f32(
                  false, a, false, b, (short)0, acc, false, false);
    }
    int mb = m0 + ((lane >> 4) << 3);   // +8 rows for upper half-wave
    #pragma unroll
    for (int r = 0; r < 8; ++r)
        Out[(size_t)(mb + r) * ldo + n0 + row] = acc[r];
}

// in_proj: [8192,96] x [384,96]^T -> [8192,384]
__global__ void k_gemm_inproj(const float* __restrict__ xn,
                              const float* __restrict__ w,
                              float* __restrict__ xz) {
    int wave = (blockIdx.x * blockDim.x + threadIdx.x) >> 5;
    int lane = threadIdx.x & 31;
    int tn = wave % 24;             // 384/16
    int tm = wave / 24;             // 8192/16 = 512
    if (tm >= 512) return;
    wmma_tile_f32(xn, C_, w, C_, xz, 2 * DI, tm * 16, tn * 16, C_, lane);
}

// out_proj: [8192,192] x [96,192]^T -> [8192,96]
__global__ void k_gemm_outproj(const float* __restrict__ yg,
                               const float* __restrict__ w,
                               float* __restrict__ yout) {
    int wave = (blockIdx.x * blockDim.x + threadIdx.x) >> 5;
    int lane = threadIdx.x & 31;
    int tn = wave % 6;              // 96/16
    int tm = wave / 6;              // 512
    if (tm >= 512) return;
    wmma_tile_f32(yg, DI, w, DI, yout, C_, tm * 16, tn * 16, DI, lane);
}

// -------------------------------------------- causal depthwise conv4 + SiLU
__global__ void k_conv_silu(const float* __restrict__ xz,
                            const float* __restrict__ cw,
                            const float* __restrict__ cb,
                            float* __restrict__ u) {
    int tid = blockIdx.x * blockDim.x + threadIdx.x;   // m*DI + d
    if (tid >= NTOK * DI) return;
    int d = tid % DI;
    int m = tid / DI;
    int l = m % L_;
    float acc = cb[d];
    #pragma unroll
    for (int j = 0; j < 4; ++j) {
        int ll = l - 3 + j;
        if (ll >= 0)
            acc += cw[d * 4 + j] * xz[(size_t)(m - l + ll) * (2 * DI) + d];
    }
    u[tid] = acc / (1.f + __expf(-acc));               // silu
}

// --------------------------------- x_proj: u[8192,192] -> dt[6], B[8], C[8]
__global__ void k_xproj(const float* __restrict__ u, const float* __restrict__ xw,
                        float* __restrict__ dtb, float* __restrict__ Bm,
                        float* __restrict__ Cm) {
    __shared__ float w[(DTR + 2 * DS) * DI];           // 22*192 floats
    for (int i = threadIdx.x; i < (DTR + 2 * DS) * DI; i += blockDim.x)
        w[i] = xw[i];
    __syncthreads();
    int m = blockIdx.x * blockDim.x + threadIdx.x;
    if (m >= NTOK) return;
    float acc[DTR + 2 * DS];
    #pragma unroll
    for (int e = 0; e < DTR + 2 * DS; ++e) acc[e] = 0.f;
    const float* ur = u + (size_t)m * DI;
    for (int d = 0; d < DI; ++d) {
        float uv = ur[d];
        #pragma unroll
        for (int e = 0; e < DTR + 2 * DS; ++e) acc[e] += uv * w[e * DI + d];
    }
    #pragma unroll
    for (int r = 0; r < DTR; ++r) dtb[(size_t)m * DTR + r] = acc[r];
    #pragma unroll
    for (int s = 0; s < DS; ++s) {
        Bm[(size_t)m * DS + s] = acc[DTR + s];
        Cm[(size_t)m * DS + s] = acc[DTR + DS + s];
    }
}

// ------------------------------------------- dt_proj + bias + softplus
__global__ void k_dtproj(const float* __restrict__ dtb, const float* __restrict__ dw,
                         const float* __restrict__ db, float* __restrict__ delta) {
    int tid = blockIdx.x * blockDim.x + threadIdx.x;   // m*DI + d
    if (tid >= NTOK * DI) return;
    int d = tid % DI;
    int m = tid / DI;
    float a = db[d];
    #pragma unroll
    for (int r = 0; r < DTR; ++r) a += dtb[(size_t)m * DTR + r] * dw[d * DTR + r];
    delta[tid] = (a > 20.f) ? a : log1pf(__expf(a));   // softplus
}

// ------------------- scan pass A: per-chunk (P = prod a, S = fold of b)
__global__ void k_scanA(const float* __restrict__ delta, const float* __restrict__ u,
                        const float* __restrict__ Bm, const float* __restrict__ A_log,
                        float* __restrict__ chP, float* __restrict__ chS) {
    int g = blockIdx.x % NCH;
    int b = blockIdx.x / NCH;
    int d = threadIdx.x;                               // 0..191
    __shared__ float sB[CHUNK * DS];
    for (int i = d; i < CHUNK * DS; i += DI)
        sB[i] = Bm[(size_t)(b * L_ + g * CHUNK) * DS + i];
    __syncthreads();
    float Av[DS], p[DS], q[DS];
    #pragma unroll
    for (int s = 0; s < DS; ++s) {
        Av[s] = -__expf(A_log[d * DS + s]);
        p[s] = 1.f; q[s] = 0.f;
    }
    int base = b * L_ + g * CHUNK;
    for (int t = 0; t < CHUNK; ++t) {
        size_t md = (size_t)(base + t) * DI + d;
        float dv = delta[md];
        float du = dv * u[md];
        #pragma unroll
        for (int s = 0; s < DS; ++s) {
            float a = __expf(dv * Av[s]);
            p[s] *= a;
            q[s] = a * q[s] + du * sB[t * DS + s];
        }
    }
    size_t o = ((size_t)(b * NCH + g) * DI + d) * DS;
    #pragma unroll
    for (int s = 0; s < DS; ++s) { chP[o + s] = p[s]; chS[o + s] = q[s]; }
}

// ------------------- scan pass B: carry chunk states (64 sequential steps)
__global__ void k_scanB(const float* __restrict__ chP, const float* __restrict__ chS,
                        float* __restrict__ Hin) {
    int tid = blockIdx.x * blockDim.x + threadIdx.x;   // b*(DI*DS) + d*DS + s
    if (tid >= B_ * DI * DS) return;
    int b   = tid / (DI * DS);
    int dss = tid % (DI * DS);
    float h = 0.f;
    for (int g = 0; g < NCH; ++g) {
        size_t o = (size_t)(b * NCH + g) * DI * DS + dss;
        Hin[o] = h;
        h = chP[o] * h + chS[o];
    }
}

// --------- scan pass C: replay chunks, y = sum_s h*C + u*D, gate by silu(z)
__global__ void k_scanC(const float* __restrict__ delta, const float* __restrict__ u,
                        const float* __restrict__ Bm, const float* __restrict__ Cm,
                        const float* __restrict__ A_log, const float* __restrict__ Hin,
                        const float* __restrict__ Dskip, const float* __restrict__ xz,
                        float* __restrict__ yg) {
    int g = blockIdx.x % NCH;
    int b = blockIdx.x / NCH;
    int d = threadIdx.x;
    __shared__ float sB[CHUNK * DS], sC[CHUNK * DS];
    for (int i = d; i < CHUNK * DS; i += DI) {
        size_t off = (size_t)(b * L_ + g * CHUNK) * DS + i;
        sB[i] = Bm[off];
        sC[i] = Cm[off];
    }
    __syncthreads();
    float Av[DS], h[DS];
    size_t ho = ((size_t)(b * NCH + g) * DI + d) * DS;
    #pragma unroll
    for (int s = 0; s < DS; ++s) {
        Av[s] = -__expf(A_log[d * DS + s]);
        h[s]  = Hin[ho + s];
    }
    float Dv = Dskip[d];
    int base = b * L_ + g * CHUNK;
    for (int t = 0; t < CHUNK; ++t) {
        size_t md = (size_t)(base + t) * DI + d;
        float dv = delta[md];
        float uv = u[md];
        float du = dv * uv;
        float y = 0.f;
        #pragma unroll
        for (int s = 0; s < DS; ++s) {
            float a = __expf(dv * Av[s]);
            h[s] = a * h[s] + du * sB[t * DS + s];
            y += h[s] * sC[t * DS + s];
        }
        y += uv * Dv;
        float zv = xz[(size_t)(base + t) * (2 * DI) + DI + d];
        y *= zv / (1.f + __expf(-zv));                 // * silu(z)
        yg[md] = y;
    }
}

// ---------------------------------------- nearest upsample x4 + residual
__global__ void k_up(const float* __restrict__ yout, const float* __restrict__ x,
                     float* __restrict__ out) {
    long i = (long)blockIdx.x * blockDim.x + threadIdx.x;
    if (i >= (long)B_ * C_ * ZD * HD * WD) return;
    int w = i % WD;
    int h = (i / WD) % HD;
    int z = (i / (WD * HD)) % ZD;
    int c = (i / ((long)WD * HD * ZD)) % C_;
    int b = i / ((long)WD * HD * ZD * C_);
    int l = (z >> 2) * 256 + (h >> 2) * 16 + (w >> 2);
    out[i] = yout[(size_t)(b * L_ + l) * C_ + c] + x[i];
}

// ============================================================================
extern "C" void kernel_launch(void* const* d_in, const int* in_sizes, int n_in,
                              void* d_out, int out_size, void* d_ws, size_t ws_size,
                              hipStream_t stream) {
    (void)in_sizes; (void)n_in; (void)out_size; (void)ws_size;
    const float* x         = (const float*)d_in[0];
    const float* ln_w      = (const float*)d_in[1];
    const float* ln_b      = (const float*)d_in[2];
    const float* in_proj_w = (const float*)d_in[3];
    const float* conv_w    = (const float*)d_in[4];
    const float* conv_b    = (const float*)d_in[5];
    const float* x_proj_w  = (const float*)d_in[6];
    const float* dt_proj_w = (const float*)d_in[7];
    const float* dt_proj_b = (const float*)d_in[8];
    const float* A_log     = (const float*)d_in[9];
    const float* D_skip    = (const float*)d_in[10];
    const float* out_proj_w= (const float*)d_in[11];
    float* out = (float*)d_out;

    float* ws    = (float*)d_ws;
    float* xn    = ws;  ws += (size_t)NTOK * C_;        // normalized tokens
    float* xz    = ws;  ws += (size_t)NTOK * 2 * DI;    // in_proj output
    float* u     = ws;  ws += (size_t)NTOK * DI;        // conv+silu
    float* dtb   = ws;  ws += (size_t)NTOK * DTR;
    float* Bm    = ws;  ws += (size_t)NTOK * DS;
    float* Cm    = ws;  ws += (size_t)NTOK * DS;
    float* delta = ws;  ws += (size_t)NTOK * DI;
    float* chP   = ws;  ws += (size_t)B_ * NCH * DI * DS;
    float* chS   = ws;  ws += (size_t)B_ * NCH * DI * DS;
    float* Hin   = ws;  ws += (size_t)B_ * NCH * DI * DS;
    float* yg    = ws;  ws += (size_t)NTOK * DI;        // gated scan output
    float* yout  = ws;  ws += (size_t)NTOK * C_;        // out_proj output

    // 1. avg-pool 4x4x4 -> tokens [B,L,C]
    k_pool<<<(B_ * C_ * L_ + 255) / 256, 256, 0, stream>>>(x, xn);
    // 2. layernorm over C (in place)
    k_ln<<<NTOK / 256, 256, 0, stream>>>(xn, ln_w, ln_b);
    // 3. in_proj GEMM (WMMA f32): 512x24 tiles, 8 waves/block
    k_gemm_inproj<<<(512 * 24) / 8, 256, 0, stream>>>(xn, in_proj_w, xz);
    // 4. causal depthwise conv + SiLU
    k_conv_silu<<<(NTOK * DI) / 256, 256, 0, stream>>>(xz, conv_w, conv_b, u);
    // 5. x_proj -> dt / B / C
    k_xproj<<<NTOK / 256, 256, 0, stream>>>(u, x_proj_w, dtb, Bm, Cm);
    // 6. dt_proj + softplus -> delta
    k_dtproj<<<(NTOK * DI) / 256, 256, 0, stream>>>(dtb, dt_proj_w, dt_proj_b, delta);
    // 7-9. chunked associative scan (64 chunks of 64)
    k_scanA<<<B_ * NCH, DI, 0, stream>>>(delta, u, Bm, A_log, chP, chS);
    k_scanB<<<(B_ * DI * DS + 255) / 256, 256, 0, stream>>>(chP, chS, Hin);
    k_scanC<<<B_ * NCH, DI, 0, stream>>>(delta, u, Bm, Cm, A_log, Hin, D_skip, xz, yg);
    // 10. out_proj GEMM (WMMA f32): 512x6 tiles
    k_gemm_outproj<<<(512 * 6) / 8, 256, 0, stream>>>(yg, out_proj_w, yout);
    // 11. upsample x4 + residual
    k_up<<<(int)(((long)B_ * C_ * ZD * HD * WD) / 256), 256, 0, stream>>>(yout, x, out);
}